// OrthogonalLinear_24086176596620
// MI455X (gfx1250) — compile-verified
//
#include <hip/hip_runtime.h>

// OrthogonalLinear via compact-WY:  out = x - ((x V) T) V^T + bias,
// V = triu(U)^T, T^{-1} = diag(u_i.u_i/2) + striu(U U^T).
// GEMMs use V_WMMA_F32_16X16X4_F32 (wave32). The two large GEMMs stage their
// B tiles in LDS with gfx1250 GLOBAL_LOAD_ASYNC_TO_LDS (ASYNCcnt) and
// double-buffer the K dimension in 64-wide chunks.

#define NMAT   1024
#define NBATCH 16384

typedef __attribute__((ext_vector_type(2))) float v2f;
typedef __attribute__((ext_vector_type(8))) float v8f;

__device__ __forceinline__ v8f wmma4(v2f a, v2f b, v8f c) {
  // (neg_a, A, neg_b, B, c_mod, C, reuse_a, reuse_b)
  return __builtin_amdgcn_wmma_f32_16x16x4_f32(false, a, false, b, (short)0, c,
                                               false, false);
}

// A fragment: 16x4 tile of row-major P at (rbase, k), row stride ld.
// ISA 7.12.2: lanes 0-15 hold K=0,1 ; lanes 16-31 hold K=2,3 -> one b64 load.
__device__ __forceinline__ v2f ldA(const float* __restrict__ P, int rbase,
                                   int k, int ld, int l, int hi) {
  return *reinterpret_cast<const v2f*>(P + (size_t)(rbase + l) * ld + k + 2 * hi);
}

// B fragment, transposed view of row-major Q (B[kk][n] = Q[cbase+n][k+kk]).
__device__ __forceinline__ v2f ldBt(const float* __restrict__ Q, int cbase,
                                    int k, int ld, int l, int hi) {
  return *reinterpret_cast<const v2f*>(Q + (size_t)(cbase + l) * ld + k + 2 * hi);
}

// B fragment, non-transposed row-major R (B[kk][n] = R[k+kk][cbase+n]).
__device__ __forceinline__ v2f ldB(const float* __restrict__ R, int cbase,
                                   int k, int ld, int l, int hi) {
  const float* p = R + (size_t)(k + 2 * hi) * ld + cbase + l;
  v2f b;
  b.x = p[0];
  b.y = p[ld];
  return b;
}

// ---------------------------------------------------- gfx1250 async helpers --

__device__ __forceinline__ void async_ld_b32(unsigned lds_off,
                                             const float* gaddr) {
  asm volatile("global_load_async_to_lds_b32 %0, %1, off"
               :: "v"(lds_off), "v"(gaddr) : "memory");
}

__device__ __forceinline__ void async_ld_b128(unsigned lds_off,
                                              const float* gaddr) {
  asm volatile("global_load_async_to_lds_b128 %0, %1, off"
               :: "v"(lds_off), "v"(gaddr) : "memory");
}

__device__ __forceinline__ void wait_async0() {
#if __has_builtin(__builtin_amdgcn_s_wait_asynccnt)
  __builtin_amdgcn_s_wait_asynccnt(0);
#else
  asm volatile("s_wait_asynccnt 0x0" ::: "memory");
#endif
}

// Low 32 bits of an LDS flat address = byte offset into the LDS allocation.
__device__ __forceinline__ unsigned lds_off32(const void* p) {
  return (unsigned)(uintptr_t)p;
}

// ---------------------------------------------------------------- kernels ---

// Ut = triu(U)
__global__ void k_mask_triu(const float* __restrict__ U, float* __restrict__ Ut) {
  int idx = blockIdx.x * blockDim.x + threadIdx.x;
  int i = idx >> 10;
  int j = idx & (NMAT - 1);
  Ut[idx] = (j >= i) ? U[idx] : 0.0f;
}

// M = T^{-1}:  upper = (Ut Ut^T), diag = (Ut Ut^T)/2, lower = 0.
__global__ void k_gram(const float* __restrict__ Ut, float* __restrict__ M) {
  int wave = (blockIdx.x * blockDim.x + threadIdx.x) >> 5;
  int lane = threadIdx.x & 31;
  int ti = wave >> 6, tj = wave & 63;          // wave-uniform
  int rbase = ti * 16, cbase = tj * 16;
  int l = lane & 15, hi = lane >> 4;
  v8f acc0 = {}, acc1 = {};
  if (tj >= ti) {                              // uniform branch: EXEC stays full
    for (int k = 0; k < NMAT; k += 8) {
      acc0 = wmma4(ldA(Ut, rbase, k,     NMAT, l, hi),
                   ldBt(Ut, cbase, k,     NMAT, l, hi), acc0);
      acc1 = wmma4(ldA(Ut, rbase, k + 4, NMAT, l, hi),
                   ldBt(Ut, cbase, k + 4, NMAT, l, hi), acc1);
    }
  }
#pragma unroll
  for (int r = 0; r < 8; ++r) {
    int row = rbase + r + 8 * hi;
    int col = cbase + l;
    float s = acc0[r] + acc1[r];
    M[(size_t)row * NMAT + col] =
        (col > row) ? s : ((col == row) ? 0.5f * s : 0.0f);
  }
}

// W = x @ Ut^T.  Block = 8 waves = 128x64 output region; wave = 16x64 strip.
// B tiles (Ut rows cbase..cbase+63, K-chunk of 64) staged K-major in LDS via
// async b32 copies (global side is transposed), double buffered.
__global__ void k_xut(const float* __restrict__ x, const float* __restrict__ Ut,
                      float* __restrict__ W) {
  __shared__ float sB[2][64 * 64];             // 2 x 16 KB, layout [kk][col]
  int tid = threadIdx.x;
  int wave = tid >> 5;
  int lane = tid & 31;
  int l = lane & 15, hi = lane >> 4;
  int rbase = (blockIdx.x >> 4) * 128 + wave * 16;  // batch rows
  int cbase = (blockIdx.x & 15) * 64;               // output column strip

  // stage chunk kc into buffer b: sB[b][kk*64 + col] = Ut[cbase+col][kc+kk]
  auto stage = [&](int b, int kc) {
#pragma unroll
    for (int i = 0; i < 16; ++i) {
      int f = i * 256 + tid;                   // 0..4095
      int col = f & 63, kk = f >> 6;
      async_ld_b32(lds_off32(&sB[b][f]),
                   Ut + (size_t)(cbase + col) * NMAT + kc + kk);
    }
  };

  v8f acc[4] = {{}, {}, {}, {}};
  stage(0, 0);
  wait_async0();
  __syncthreads();

  for (int c = 0; c < NMAT / 64; ++c) {
    if (c + 1 < NMAT / 64) stage((c + 1) & 1, (c + 1) * 64);
    const float* B = &sB[c & 1][0];
    for (int k4 = 0; k4 < 64; k4 += 4) {
      v2f a = ldA(x, rbase, c * 64 + k4, NMAT, l, hi);
#pragma unroll
      for (int t = 0; t < 4; ++t) {
        v2f b;
        b.x = B[(k4 + 2 * hi)     * 64 + 16 * t + l];
        b.y = B[(k4 + 2 * hi + 1) * 64 + 16 * t + l];
        acc[t] = wmma4(a, b, acc[t]);
      }
    }
    if (c + 1 < NMAT / 64) {
      wait_async0();
      __syncthreads();
    }
  }

#pragma unroll
  for (int t = 0; t < 4; ++t)
#pragma unroll
    for (int r = 0; r < 8; ++r) {
      int row = rbase + r + 8 * hi;
      W[(size_t)row * NMAT + cbase + 16 * t + l] = acc[t][r];
    }
}

// Column-block update of the solve Y M = W (Y in place in W):
// W[:, jb*64 ..] -= W[:, 0..64*jb-1] @ M[0..64*jb-1, jb*64..]
__global__ void k_update(float* __restrict__ W, const float* __restrict__ M,
                         int jblock) {
  int wave = (blockIdx.x * blockDim.x + threadIdx.x) >> 5;
  int lane = threadIdx.x & 31;
  int bi = wave >> 2;
  int tj = wave & 3;
  int rbase = bi * 16;
  int cbase = jblock * 64 + tj * 16;
  int Klen = jblock * 64;
  int l = lane & 15, hi = lane >> 4;
  v8f acc0 = {}, acc1 = {};
  for (int k = 0; k < Klen; k += 8) {
    acc0 = wmma4(ldA(W, rbase, k,     NMAT, l, hi),
                 ldB(M, cbase, k,     NMAT, l, hi), acc0);
    acc1 = wmma4(ldA(W, rbase, k + 4, NMAT, l, hi),
                 ldB(M, cbase, k + 4, NMAT, l, hi), acc1);
  }
#pragma unroll
  for (int r = 0; r < 8; ++r) {
    int row = rbase + r + 8 * hi;
    W[(size_t)row * NMAT + cbase + l] -= acc0[r] + acc1[r];
  }
}

// Per-row forward substitution against diagonal block M_jj (LDS-staged).
__global__ void k_trisolve(float* __restrict__ W, const float* __restrict__ M,
                           int jblock) {
  __shared__ float Mjj[64 * 64];
  int base = jblock * 64;
  for (int t = threadIdx.x; t < 64 * 64; t += blockDim.x)
    Mjj[t] = M[(size_t)(base + (t >> 6)) * NMAT + base + (t & 63)];
  __syncthreads();

  int row = blockIdx.x * blockDim.x + threadIdx.x;
  float* wrow = W + (size_t)row * NMAT + base;
  float y[64];
#pragma unroll 8
  for (int c = 0; c < 64; ++c) y[c] = wrow[c];
  for (int c = 0; c < 64; ++c) {
    float s = y[c];
    for (int cp = 0; cp < c; ++cp) s -= y[cp] * Mjj[cp * 64 + c];
    y[c] = s / Mjj[c * 64 + c];
  }
#pragma unroll 8
  for (int c = 0; c < 64; ++c) wrow[c] = y[c];
}

// out = x - Y @ Ut + bias.  Same blocking as k_xut; Ut tile is K-major in
// memory already, so async staging uses contiguous b128 copies.
__global__ void k_final(const float* __restrict__ Y, const float* __restrict__ Ut,
                        const float* __restrict__ x, const float* __restrict__ bias,
                        float* __restrict__ out) {
  __shared__ float sB[2][64 * 64];             // layout [kk][col]
  int tid = threadIdx.x;
  int wave = tid >> 5;
  int lane = tid & 31;
  int l = lane & 15, hi = lane >> 4;
  int rbase = (blockIdx.x >> 4) * 128 + wave * 16;
  int cbase = (blockIdx.x & 15) * 64;

  // sB[b][kk*64 + col] = Ut[kc+kk][cbase+col]  (contiguous both sides)
  auto stage = [&](int b, int kc) {
#pragma unroll
    for (int i = 0; i < 4; ++i) {
      int f = i * 1024 + tid * 4;              // 0..4092, step 4
      int kk = f >> 6, col = f & 63;
      async_ld_b128(lds_off32(&sB[b][f]),
                    Ut + (size_t)(kc + kk) * NMAT + cbase + col);
    }
  };

  v8f acc[4] = {{}, {}, {}, {}};
  stage(0, 0);
  wait_async0();
  __syncthreads();

  for (int c = 0; c < NMAT / 64; ++c) {
    if (c + 1 < NMAT / 64) stage((c + 1) & 1, (c + 1) * 64);
    const float* B = &sB[c & 1][0];
    for (int k4 = 0; k4 < 64; k4 += 4) {
      v2f a = ldA(Y, rbase, c * 64 + k4, NMAT, l, hi);
#pragma unroll
      for (int t = 0; t < 4; ++t) {
        v2f b;
        b.x = B[(k4 + 2 * hi)     * 64 + 16 * t + l];
        b.y = B[(k4 + 2 * hi + 1) * 64 + 16 * t + l];
        acc[t] = wmma4(a, b, acc[t]);
      }
    }
    if (c + 1 < NMAT / 64) {
      wait_async0();
      __syncthreads();
    }
  }

#pragma unroll
  for (int t = 0; t < 4; ++t) {
    int col = cbase + 16 * t + l;
    float bb = bias[col];
#pragma unroll
    for (int r = 0; r < 8; ++r) {
      int row = rbase + r + 8 * hi;
      size_t idx = (size_t)row * NMAT + col;
      out[idx] = x[idx] - acc[t][r] + bb;
    }
  }
}

// ------------------------------------------------------------------ launch ---

extern "C" void kernel_launch(void* const* d_in, const int* in_sizes, int n_in,
                              void* d_out, int out_size, void* d_ws, size_t ws_size,
                              hipStream_t stream) {
  const float* x    = (const float*)d_in[0];   // [16384,1024]
  const float* U    = (const float*)d_in[1];   // [1024,1024]
  const float* bias = (const float*)d_in[2];   // [1,1024]
  float* out = (float*)d_out;

  float* Ut = (float*)d_ws;                    //  4 MB
  float* M  = Ut + (size_t)NMAT * NMAT;        //  4 MB
  float* W  = M  + (size_t)NMAT * NMAT;        // 64 MB (W, then Y in place)

  // 1) mask upper triangle
  k_mask_triu<<<(NMAT * NMAT) / 256, 256, 0, stream>>>(U, Ut);

  // 2) M = T^{-1} from Gram matrix (64x64 tiles, 8 waves/block)
  k_gram<<<(64 * 64) / 8, 256, 0, stream>>>(Ut, M);

  // 3) W = x @ Ut^T  (128x64 per block, async LDS B-tiles)
  k_xut<<<(NBATCH / 128) * (NMAT / 64), 256, 0, stream>>>(x, Ut, W);

  // 4) blocked triangular solve  Y M = W  (in place), 16 blocks of 64 cols
  for (int j = 0; j < 16; ++j) {
    if (j > 0) k_update<<<(1024 * 4) / 8, 256, 0, stream>>>(W, M, j);
    k_trisolve<<<NBATCH / 256, 256, 0, stream>>>(W, M, j);
  }

  // 5) out = x - Y @ Ut + bias
  k_final<<<(NBATCH / 128) * (NMAT / 64), 256, 0, stream>>>(W, Ut, x, bias, out);
}